// Fssn_layers_19267223290399
// MI455X (gfx1250) — compile-verified
//
#include <hip/hip_runtime.h>
#include <float.h>

typedef __attribute__((ext_vector_type(2))) float v2f;
typedef __attribute__((ext_vector_type(8))) float v8f;

#define LEAKY_ALPHA 0.2f

// ---------------------------------------------------------------------------
// Kernel A: fill aggregation buffer (== d_out) with -FLT_MAX
// ---------------------------------------------------------------------------
__global__ void fill_kernel(float* __restrict__ out, long n4) {
  long t = (long)blockIdx.x * blockDim.x + threadIdx.x;
  if (t < n4) {
    float4 v = make_float4(-FLT_MAX, -FLT_MAX, -FLT_MAX, -FLT_MAX);
    ((float4*)out)[t] = v;
  }
}

// ---------------------------------------------------------------------------
// Kernel B: per-edge attention matmul via V_WMMA_F32_16X16X4_F32 (two edges
// K-packed per WMMA) + full-wave scatter-max with global_atomic_max_num_f32.
//
// One wave handles one (type-block i, edge-pair {e0,e1}).
//   A (16x4):  rows 0-3  = e0 heads (k0,k1 = its att weights, k2,k3 = 0)
//              rows 8-11 = e1 heads (k0,k1 = 0, k2,k3 = its att weights)
//              other rows = 0
//   B (4x16):  k0,k1 = e0's two gathered feature rows (lanes 0-15),
//              k2,k3 = e1's (lanes 16-31); 16 d-columns per WMMA, 8 WMMAs.
//   D (16x16): VGPR j (j=0..3): lanes 0-15 hold M=j   (e0 head j),
//                               lanes 16-31 hold M=j+8 (e1 head j)
//   -> every lane scatters its own edge's 4 head values: no divergence,
//      one base address per lane, per-(head,slice) displacement is an
//      immediate `offset:` on the atomic.
//   E odd: e1=e0 duplicate; duplicate atomics are idempotent under max.
// ---------------------------------------------------------------------------
__global__ __launch_bounds__(256) void edge_kernel(
    const int* __restrict__ batch,     // [E, 3] int32
    const float* __restrict__ bf,      // [N, 128] f32
    const float* __restrict__ aw,      // [3E, 4, 2] f32
    float* __restrict__ agg,           // [N, 4, 128] f32 (== d_out)
    int E) {
  const int lane = threadIdx.x & 31;
  const long wave = ((long)blockIdx.x * blockDim.x + threadIdx.x) >> 5;
  const int pairs = (E + 1) >> 1;
  if (wave >= 3L * pairs) return;                 // wave-uniform exit
  const int i = (int)(wave / pairs);              // type block 0..2
  const int p = (int)(wave - (long)i * pairs);
  const int e0 = 2 * p;
  int e1 = e0 + 1;
  if (e1 >= E) e1 = e0;                           // duplicate: max-idempotent

  const int half = lane >> 4;                     // 0 -> edge e0, 1 -> edge e1
  const int lh   = lane & 15;
  // the two "other" columns, in ascending order (matches reference concat)
  const int c0 = (i == 0) ? 1 : 0;
  const int c1 = (i == 2) ? 1 : 2;
  const int eMine = half ? e1 : e0;

  const int node = batch[(size_t)eMine * 3 + i];  // this edge's scatter target
  const int n0   = batch[(size_t)eMine * 3 + c0];
  const int n1   = batch[(size_t)eMine * 3 + c1];
  const float* __restrict__ f0 = bf + (size_t)n0 * 128;
  const float* __restrict__ f1 = bf + (size_t)n1 * 128;

  // A fragment: lane l supplies row M=lh; K-pair (k0,k1) for l<16, (k2,k3)
  // for l>=16.  Nonzero rows: 0-3 from e0 (first half), 8-11 from e1
  // (second half) -> exact zero-padding elsewhere.
  v2f a; a.x = 0.0f; a.y = 0.0f;
  const bool act = half ? (lh >= 8 && lh < 12) : (lh < 4);
  if (act) {
    const int head = lh & 3;
    const v2f* wp = (const v2f*)(aw + (((size_t)i * E + eMine) * 4 + head) * 2);
    a = *wp;
  }

  // Preload all B fragments (one load clause, L2-resident gathers)
  float b0[8], b1[8];
#pragma unroll
  for (int s = 0; s < 8; ++s) {
    b0[s] = f0[s * 16 + lh];
    b1[s] = f1[s * 16 + lh];
  }

  // One base address per lane; head/slice displacement is an immediate.
  float* base = agg + ((size_t)node << 9) + lh;   // node*H*D + d%16

#pragma unroll
  for (int s = 0; s < 8; ++s) {
    v2f b; b.x = b0[s]; b.y = b1[s];
    v8f c = {};
    v8f acc = __builtin_amdgcn_wmma_f32_16x16x4_f32(
        false, a, false, b, (short)0, c, false, false);
#pragma unroll
    for (int j = 0; j < 4; ++j) {                 // head j, VGPR j of D
      float v = acc[j];
      asm volatile(
          "global_atomic_max_num_f32 %0, %1, off offset:%2 scope:SCOPE_DEV"
          :: "v"(base), "v"(v), "n"((j * 128 + s * 16) * 4)
          : "memory");
    }
  }
}

// ---------------------------------------------------------------------------
// Kernel C: out = leaky_relu(batch_features broadcast + aggregated), in place
// ---------------------------------------------------------------------------
__global__ void finalize_kernel(const float* __restrict__ bf,
                                float* __restrict__ out, long n4) {
  long t = (long)blockIdx.x * blockDim.x + threadIdx.x;
  if (t >= n4) return;
  long f = t << 2;                 // flat float index into [N, 4, 128]
  int n = (int)(f >> 9);           // / 512
  int d = (int)(f & 127);          // within D (multiple of 4)
  float4 agg = ((const float4*)out)[t];
  float4 b = *(const float4*)(bf + (size_t)n * 128 + d);
  float4 r;
  float x;
  x = agg.x + b.x; r.x = (x > 0.0f) ? x : LEAKY_ALPHA * x;
  x = agg.y + b.y; r.y = (x > 0.0f) ? x : LEAKY_ALPHA * x;
  x = agg.z + b.z; r.z = (x > 0.0f) ? x : LEAKY_ALPHA * x;
  x = agg.w + b.w; r.w = (x > 0.0f) ? x : LEAKY_ALPHA * x;
  ((float4*)out)[t] = r;
}

// ---------------------------------------------------------------------------
extern "C" void kernel_launch(void* const* d_in, const int* in_sizes, int n_in,
                              void* d_out, int out_size, void* d_ws, size_t ws_size,
                              hipStream_t stream) {
  (void)n_in; (void)d_ws; (void)ws_size;
  const int*   batch = (const int*)d_in[0];    // [E, 3]
  const float* bf    = (const float*)d_in[1];  // [N, 128]
  const float* aw    = (const float*)d_in[2];  // [3E, 4, 2]
  float* out = (float*)d_out;                  // [N, 512]

  const int E = in_sizes[0] / 3;
  const long total = (long)out_size;           // N*H*D, divisible by 4
  const long n4 = total >> 2;

  dim3 blk(256);
  fill_kernel<<<(unsigned)((n4 + 255) / 256), blk, 0, stream>>>(out, n4);

  const long waves = 3L * ((E + 1) / 2);
  const long thr = waves * 32;
  edge_kernel<<<(unsigned)((thr + 255) / 256), blk, 0, stream>>>(
      batch, bf, aw, out, E);

  finalize_kernel<<<(unsigned)((n4 + 255) / 256), blk, 0, stream>>>(bf, out, n4);
}